// robotDHLearnable_14903536517848
// MI455X (gfx1250) — compile-verified
//
#include <hip/hip_runtime.h>

#define TPB 256

typedef float v2f __attribute__((ext_vector_type(2)));
typedef float v8f __attribute__((ext_vector_type(8)));

// ---------------------------------------------------------------------------
// Prep: pack per-joint constants [sign, offset, d, a, sin(alpha), cos(alpha)]
// (7 x 6 floats) plus the 4x4 base matrix (16 floats) into d_ws.
// ---------------------------------------------------------------------------
__global__ void fk_prep(const float* __restrict__ sgn, const float* __restrict__ off,
                        const float* __restrict__ dd, const float* __restrict__ aa,
                        const float* __restrict__ alpha, const float* __restrict__ base,
                        float* __restrict__ ws) {
    int j = threadIdx.x;
    if (j < 7) {
        float s, c;
        __sincosf(alpha[j], &s, &c);
        ws[j * 6 + 0] = sgn[j];
        ws[j * 6 + 1] = off[j];
        ws[j * 6 + 2] = dd[j];
        ws[j * 6 + 3] = aa[j];
        ws[j * 6 + 4] = s;
        ws[j * 6 + 5] = c;
    }
    if (j < 16) ws[42 + j] = base[j];
}

// ---------------------------------------------------------------------------
// Main: one thread per timestep.
//   in  : angles staged per-wave into LDS via global_load_async_to_lds_b128
//   math: P = T0*...*T6 with a 3x4 accumulator (DH transforms keep row3 =
//         [0,0,0,1], closed under multiplication). Final out = base*P done on
//         the matrix pipe: V_WMMA_F32_16X16X4_F32 with A=[base x4] (constant,
//         built from uniforms) and B=[P_t0|P_t1|P_t2|P_t3] from LDS — one
//         wmma per 4 timesteps, offloading 48 FMAs/wave from the VALU.
//   out : results staged to LDS, streamed via global_store_async_from_lds_b128
//         (each instruction writes a fully contiguous 512B per wave)
// ---------------------------------------------------------------------------
__global__ __launch_bounds__(TPB) void fk_main(
    const float* __restrict__ angles, const float* __restrict__ base,
    const float* __restrict__ sgn, const float* __restrict__ off,
    const float* __restrict__ dd, const float* __restrict__ aa,
    const float* __restrict__ alpha, const float* __restrict__ packed,
    float* __restrict__ out, int ts)
{
    __shared__ float sang[TPB * 7];   //  7168 B : block's angle slice
    __shared__ float sp[TPB * 16];    // 16384 B : per-timestep P matrices (WMMA B source)
    __shared__ float sout[TPB * 16];  // 16384 B : result staging for async store

    const int tid = threadIdx.x;
    const int block_start = blockIdx.x * TPB;
    const int t = block_start + tid;

    // ---- per-joint constants (uniform -> scalar loads) ----
    float cs[7], co[7], cd[7], caj[7], csa[7], cca[7], b[16];
    if (packed) {
#pragma unroll
        for (int j = 0; j < 7; j++) {
            cs[j]  = packed[j * 6 + 0];
            co[j]  = packed[j * 6 + 1];
            cd[j]  = packed[j * 6 + 2];
            caj[j] = packed[j * 6 + 3];
            csa[j] = packed[j * 6 + 4];
            cca[j] = packed[j * 6 + 5];
        }
#pragma unroll
        for (int i = 0; i < 16; i++) b[i] = packed[42 + i];
    } else {
#pragma unroll
        for (int j = 0; j < 7; j++) {
            cs[j]  = sgn[j];
            co[j]  = off[j];
            cd[j]  = dd[j];
            caj[j] = aa[j];
            float s, c;
            __sincosf(alpha[j], &s, &c);
            csa[j] = s;
            cca[j] = c;
        }
#pragma unroll
        for (int i = 0; i < 16; i++) b[i] = base[i];
    }

    // ---- stage this wave's 32*7 floats (896 B = 56 x 16B chunks) into LDS ----
    float ang[7];
    const bool full = (block_start + TPB) <= ts;
    const unsigned lane = tid & 31u;
    if (full) {
        const unsigned wave_b = (unsigned)(tid & ~31) * 28u;      // byte offset of wave slice
        const char* g0 = (const char*)angles + (size_t)block_start * 28u + wave_b + lane * 16u;
        const unsigned lds_base = (unsigned)(size_t)(&sang[0]);   // low 32 bits = LDS offset
        const unsigned l0 = lds_base + wave_b + lane * 16u;
        asm volatile("global_load_async_to_lds_b128 %0, %1, off"
                     :: "v"(l0), "v"(g0) : "memory");
        if (lane < 24u) {  // chunks 32..55 cover bytes 512..896
            const char* g1 = g0 + 512;
            const unsigned l1 = l0 + 512u;
            asm volatile("global_load_async_to_lds_b128 %0, %1, off"
                         :: "v"(l1), "v"(g1) : "memory");
        }
        asm volatile("s_wait_asynccnt 0x0" ::: "memory");
#pragma unroll
        for (int j = 0; j < 7; j++) ang[j] = sang[tid * 7 + j];
    } else {
        if (t < ts) {
#pragma unroll
            for (int j = 0; j < 7; j++) ang[j] = angles[(size_t)t * 7 + j];
        }
    }

    if (!full && t >= ts) return;

    // ---- P = T0 (3x4 accumulator; row3 == [0,0,0,1] implicitly) ----
    float p[3][4];
    {
        const float th0 = __fmaf_rn(cs[0], ang[0], co[0]);
        float st, ct;
        __sincosf(th0, &st, &ct);
        const float sa = csa[0], ca = cca[0], A = caj[0];
        p[0][0] = ct;  p[0][1] = -st * ca; p[0][2] = st * sa;  p[0][3] = A * ct;
        p[1][0] = st;  p[1][1] = ct * ca;  p[1][2] = -ct * sa; p[1][3] = A * st;
        p[2][0] = 0.f; p[2][1] = sa;       p[2][2] = ca;       p[2][3] = cd[0];
    }

    // ---- chain: P = P * T_j for j=1..6, rows independent, row3 stays fixed ----
    // T columns: c0=[ct,st,0,0] c1=[-st*ca,ct*ca,sa,0] c2=[st*sa,-ct*sa,ca,0] c3=[a*ct,a*st,d,1]
#pragma unroll
    for (int j = 1; j < 7; j++) {
        const float th = __fmaf_rn(cs[j], ang[j], co[j]);
        float st, ct;
        __sincosf(th, &st, &ct);
        const float sa = csa[j], ca = cca[j], A = caj[j], D = cd[j];
#pragma unroll
        for (int i = 0; i < 3; i++) {
            const float r0 = p[i][0], r1 = p[i][1], r2 = p[i][2], r3 = p[i][3];
            const float u = __fmaf_rn(r0, ct, r1 * st);      // new col0
            const float v = __fmaf_rn(r1, ct, -(r0 * st));
            p[i][0] = u;
            p[i][1] = __fmaf_rn(v, ca, r2 * sa);             // v*ca + r2*sa
            p[i][2] = __fmaf_rn(r2, ca, -(v * sa));          // -v*sa + r2*ca
            p[i][3] = __fmaf_rn(A, u, __fmaf_rn(D, r2, r3)); // a*u + d*r2 + r3
        }
    }

    if (full) {
        // ================= out = base * P on the matrix pipe =================
        // Stage P (incl. implicit row3) into LDS as the WMMA B source.
        float4* pp = (float4*)(sp + (size_t)tid * 16);
        pp[0] = make_float4(p[0][0], p[0][1], p[0][2], p[0][3]);
        pp[1] = make_float4(p[1][0], p[1][1], p[1][2], p[1][3]);
        pp[2] = make_float4(p[2][0], p[2][1], p[2][2], p[2][3]);
        pp[3] = make_float4(0.f, 0.f, 0.f, 1.f);

        // A = [base;base;base;base] in 16x4 f32 A-layout:
        //   lane m: VGPR0 = A[m][K], lanes 0-15 K=0, lanes 16-31 K=2; VGPR1: K=1/K=3
        const int  row = (int)(lane & 3u);     // A row m mod 4 (base row)
        const unsigned q = lane & 15u;
        const unsigned h = lane >> 4;          // 0: K0/K1, rows 0-1 ; 1: K2/K3, rows 2-3
        float ak0 = row == 1 ? b[4]  : b[0];  ak0 = row == 2 ? b[8]  : ak0;  ak0 = row == 3 ? b[12] : ak0;
        float ak1 = row == 1 ? b[5]  : b[1];  ak1 = row == 2 ? b[9]  : ak1;  ak1 = row == 3 ? b[13] : ak1;
        float ak2 = row == 1 ? b[6]  : b[2];  ak2 = row == 2 ? b[10] : ak2;  ak2 = row == 3 ? b[14] : ak2;
        float ak3 = row == 1 ? b[7]  : b[3];  ak3 = row == 2 ? b[11] : ak3;  ak3 = row == 3 ? b[15] : ak3;
        v2f A;
        A.x = h ? ak2 : ak0;
        A.y = h ? ak3 : ak1;

        const v8f cz = {0.f, 0.f, 0.f, 0.f, 0.f, 0.f, 0.f, 0.f};
        const int wl   = tid & ~31;            // wave's first local timestep
        const int patt = (int)(h * 8u + (q & 3u));  // (2h)-th row *4 + col

        // 8 tiles x 4 timesteps: B = [P_t0|P_t1|P_t2|P_t3] (4x16), D = A*B.
        // D row-blocks are redundant (A replicates mod 4): lanes 0-15 == 16-31,
        // so lane halves store rows {0,1} / {2,3} of their timestep's result.
#pragma unroll
        for (int g = 0; g < 8; g++) {
            const int tb = (wl + g * 4 + (int)(q >> 2)) * 16;
            v2f B;
            B.x = sp[tb + patt];          // B rows 0 / 2 (lane halves)
            B.y = sp[tb + patt + 4];      // B rows 1 / 3
            v8f dacc = __builtin_amdgcn_wmma_f32_16x16x4_f32(
                false, A, false, B, (short)0, cz, false, false);
            sout[tb + patt]     = h ? dacc[2] : dacc[0];  // result row 2h
            sout[tb + patt + 4] = h ? dacc[3] : dacc[1];  // result row 2h+1
        }

        asm volatile("s_wait_dscnt 0x0" ::: "memory");  // LDS staging visible to async engine

        const unsigned wave_lds = (unsigned)(size_t)(&sout[0]) + (unsigned)(tid & ~31) * 64u;
        char* gout = (char*)out + ((size_t)block_start + (unsigned)(tid & ~31)) * 64u;
#pragma unroll
        for (int k = 0; k < 4; k++) {
            const unsigned ls = wave_lds + (unsigned)k * 512u + lane * 16u;
            char* ga = gout + (size_t)k * 512u + lane * 16u;
            asm volatile("global_store_async_from_lds_b128 %0, %1, off"
                         :: "v"(ga), "v"(ls) : "memory");
        }
        // No explicit s_wait_asynccnt: S_ENDPGM performs an implicit wait-idle,
        // so the async stores complete before this wave's LDS can be freed.
    } else {
        // ---- tail: out = base * P on the VALU (P row3 = [0,0,0,1]) ----
        float r[4][4];
#pragma unroll
        for (int i = 0; i < 4; i++) {
            const float b0 = b[i * 4 + 0], b1 = b[i * 4 + 1], b2 = b[i * 4 + 2], b3 = b[i * 4 + 3];
#pragma unroll
            for (int c = 0; c < 3; c++)
                r[i][c] = __fmaf_rn(b0, p[0][c], __fmaf_rn(b1, p[1][c], b2 * p[2][c]));
            r[i][3] = __fmaf_rn(b0, p[0][3], __fmaf_rn(b1, p[1][3], __fmaf_rn(b2, p[2][3], b3)));
        }
        float4* o = (float4*)(out + (size_t)t * 16);
#pragma unroll
        for (int i = 0; i < 4; i++)
            o[i] = make_float4(r[i][0], r[i][1], r[i][2], r[i][3]);
    }
}

extern "C" void kernel_launch(void* const* d_in, const int* in_sizes, int n_in,
                              void* d_out, int out_size, void* d_ws, size_t ws_size,
                              hipStream_t stream) {
    const float* base   = (const float*)d_in[0];
    const float* angles = (const float*)d_in[1];
    const float* sgn    = (const float*)d_in[2];
    const float* off    = (const float*)d_in[3];
    const float* dd     = (const float*)d_in[4];
    const float* aa     = (const float*)d_in[5];
    const float* alpha  = (const float*)d_in[6];
    float* out = (float*)d_out;

    const int ts = in_sizes[1] / 7;
    const bool use_ws = (ws_size >= 58 * sizeof(float));
    float* ws = (float*)d_ws;

    if (use_ws)
        fk_prep<<<1, 32, 0, stream>>>(sgn, off, dd, aa, alpha, base, ws);

    const int grid = (ts + TPB - 1) / TPB;
    fk_main<<<grid, TPB, 0, stream>>>(angles, base, sgn, off, dd, aa, alpha,
                                      use_ws ? ws : nullptr, out, ts);
}